// RecurrentNetwork_3547642986523
// MI455X (gfx1250) — compile-verified
//
#include <hip/hip_runtime.h>

typedef __attribute__((ext_vector_type(2))) float v2f;
typedef __attribute__((ext_vector_type(8))) float v8f;

#if defined(__has_builtin)
#if __has_builtin(__builtin_amdgcn_global_load_async_to_lds_b32) && \
    __has_builtin(__builtin_amdgcn_s_wait_asynccnt)
#define USE_ASYNC_LDS 1
#endif
#endif
#ifndef USE_ASYNC_LDS
#define USE_ASYNC_LDS 0
#endif

typedef __attribute__((address_space(1))) int g_i32;
typedef __attribute__((address_space(3))) int l_i32;

namespace {
constexpr int   kTSteps           = 500000;
constexpr int   kSegLen           = 100;                       // steps per replay thread
constexpr int   kNSeg             = kTSteps / kSegLen;         // 5000
constexpr int   kSegsPerBlock     = 125;                       // replay segments per block
constexpr int   kNBlocks          = kNSeg / kSegsPerBlock;     // 40
constexpr int   kSegsPerGemmBlock = 64;                        // 4 waves x 16 cols
constexpr int   kGemmBlocks       = (kNSeg + kSegsPerGemmBlock - 1) / kSegsPerGemmBlock; // 79
constexpr float kDt               = 0.001f;
constexpr float kOmegaF           = 6.283185f;

struct Params {
  double TL[16];   // T^kSegLen, row-major 4x4
  double TB[16];   // TL^kSegsPerBlock
  float  M[4];     // x-transition I + dt*(A - xicov C), row-major 2x2
  float  F[4];     // f-transition I + dt*K
  float  g[2];     // xicov column 0 (only dy[0] couples in)
  float  cov[3];   // vx, vp, cxp steady-state (CARE fixed point)
  float  sdt;      // sqrt(4*eta*kappa)*dt  for dy_hat
};
} // namespace

// ---------------------------------------------------------------------------
// Phase 0: single-thread setup. The reference covariance starts at the CARE
// fixed point, where the Riccati Euler increment is identically zero, so
// vx/vp/cxp are trajectory constants. Solve the CARE (3 scalar equations,
// double), build the f32 per-step matrices, the double 4x4 segment/block
// transition powers, and the per-segment response coefficients
// Coef[:,j] = M^(kSegLen-1-j) g.
// ---------------------------------------------------------------------------
__global__ void k_setup(const float* __restrict__ Kin, Params* __restrict__ P,
                        float2* __restrict__ coefs) {
  if (threadIdx.x != 0 || blockIdx.x != 0) return;
  const double gam = 0.3, om = 6.283185;
  // CARE:  A X + X A^T + D - X C^T C X = 0,  D = 1.75 I, C^T C = 3.2 e00.
  double a = 0.7, b = 0.7, c = 0.1;
  for (int i = 0; i < 400; ++i) {
    c = (-2.0 * om + sqrt(4.0 * om * om - 12.8 * (gam * b - 1.75))) / 6.4;
    a = (-gam + sqrt(gam * gam + 12.8 * (2.0 * om * c + 1.75))) / 6.4;
    b = a + c * (gam + 3.2 * a) / om;
  }
  const float vx = (float)a, vp = (float)b, cxp = (float)c;
  const float s = sqrtf(3.2f);

  const float M00 = 1.0f + kDt * (-0.15f - 3.2f * vx);
  const float M01 = kDt * kOmegaF;
  const float M10 = kDt * (-kOmegaF - 3.2f * cxp);
  const float M11 = 1.0f - kDt * 0.15f;
  const float g0 = s * vx, g1 = s * cxp;
  const float F00 = 1.0f + kDt * Kin[0], F01 = kDt * Kin[1];
  const float F10 = kDt * Kin[2],        F11 = 1.0f + kDt * Kin[3];

  P->M[0] = M00; P->M[1] = M01; P->M[2] = M10; P->M[3] = M11;
  P->F[0] = F00; P->F[1] = F01; P->F[2] = F10; P->F[3] = F11;
  P->g[0] = g0;  P->g[1] = g1;
  P->cov[0] = vx; P->cov[1] = vp; P->cov[2] = cxp;
  P->sdt = s * kDt;

  // z = [x0 x1 f0 f1]; z' = T z + g*dy0
  double T[16] = { (double)M00, (double)M01, 0.0, 0.0,
                   (double)M10, (double)M11, (double)kDt * F00, (double)kDt * F01,
                   0.0, 0.0, (double)F00, (double)F01,
                   0.0, 0.0, (double)F10, (double)F11 };
  double TL[16] = {1,0,0,0, 0,1,0,0, 0,0,1,0, 0,0,0,1};
  for (int it = 0; it < kSegLen; ++it) {
    double N[16];
    for (int r = 0; r < 4; ++r)
      for (int cc = 0; cc < 4; ++cc) {
        double acc = 0.0;
        for (int k = 0; k < 4; ++k) acc += T[r*4+k] * TL[k*4+cc];
        N[r*4+cc] = acc;
      }
    for (int i = 0; i < 16; ++i) TL[i] = N[i];
  }
  double TB[16] = {1,0,0,0, 0,1,0,0, 0,0,1,0, 0,0,0,1};
  for (int it = 0; it < kSegsPerBlock; ++it) {
    double N[16];
    for (int r = 0; r < 4; ++r)
      for (int cc = 0; cc < 4; ++cc) {
        double acc = 0.0;
        for (int k = 0; k < 4; ++k) acc += TL[r*4+k] * TB[k*4+cc];
        N[r*4+cc] = acc;
      }
    for (int i = 0; i < 16; ++i) TB[i] = N[i];
  }
  for (int i = 0; i < 16; ++i) { P->TL[i] = TL[i]; P->TB[i] = TB[i]; }

  // Coef[:,kSegLen-1] = g; Coef[:,j] = M Coef[:,j+1]
  double u0 = (double)g0, u1 = (double)g1;
  for (int j = kSegLen - 1; j >= 0; --j) {
    coefs[j] = make_float2((float)u0, (float)u1);
    const double n0 = (double)M00 * u0 + (double)M01 * u1;
    const double n1 = (double)M10 * u0 + (double)M11 * u1;
    u0 = n0; u1 = n1;
  }
}

// ---------------------------------------------------------------------------
// Phase 1: R(2 x 5000) = CoefPad(16 x 100) @ DY0(100 x 5000) via
// v_wmma_f32_16x16x4_f32. Each wave owns 16 segments (one 16x16 D tile,
// rows 0..1 live). dy0 is staged to LDS (async global->LDS when available).
// The padded A matrix is pre-expanded in LDS as float2 K-pairs so the inner
// loop is exactly: ds_load_b64 (A) + ds_load_b64 (B) + wmma — branchless,
// selectless, EXEC untouched (WMMA requires EXEC all-ones).
// ---------------------------------------------------------------------------
__global__ void __launch_bounds__(128)
k_resp(const float* __restrict__ dys,
       const float2* __restrict__ coefs,
       float2* __restrict__ R) {
  __shared__ float  sdy[kSegsPerGemmBlock * kSegLen];   // 25.6 KB, zero-padded tail
  __shared__ float2 scoA2[(kSegLen / 2) * 16];          // A pairs: [kpair][row], 6.4 KB

  const int tid = threadIdx.x;
  const int segBase = blockIdx.x * kSegsPerGemmBlock;
  const int nsegHere = min(kSegsPerGemmBlock, kNSeg - segBase);
  const int nsteps = nsegHere * kSegLen;
  const long base = (long)segBase * kSegLen;

  // Pre-expand A: scoA2[kp*16 + row] = {Coef[row][2kp], Coef[row][2kp+1]},
  // rows 2..15 zero. Branches here are outside the hot loop.
  for (int i = tid; i < (kSegLen / 2) * 16; i += blockDim.x) {
    const int kp = i >> 4, row = i & 15;
    float2 v = make_float2(0.0f, 0.0f);
    if (row == 0) v = make_float2(coefs[2 * kp].x, coefs[2 * kp + 1].x);
    else if (row == 1) v = make_float2(coefs[2 * kp].y, coefs[2 * kp + 1].y);
    scoA2[i] = v;
  }
  // zero-fill the dy tail so the WMMA loop needs no bounds checks
  for (int i = nsteps + tid; i < kSegsPerGemmBlock * kSegLen; i += blockDim.x)
    sdy[i] = 0.0f;
#if USE_ASYNC_LDS
  for (int i = tid; i < nsteps; i += blockDim.x) {
    __builtin_amdgcn_global_load_async_to_lds_b32(
        (g_i32*)(dys + 2 * (base + i)),
        (l_i32*)(&sdy[i]), 0, 0);
  }
  __builtin_amdgcn_s_wait_asynccnt(0);
#else
  for (int i = tid; i < nsteps; i += blockDim.x)
    sdy[i] = dys[2 * (base + i)];   // dys is (T,2); only dy[0] couples
#endif
  __syncthreads();

  const int wave = tid >> 5;
  const int lane = tid & 31;
  const int s0 = wave * 16;            // this wave's first local segment
  if (s0 >= nsegHere) return;          // wave-uniform tail cut
  const int col = lane & 15;           // A row index == B/D column index
  const int khalf = (lane >> 4) * 2;   // K sub-block held by upper lane half
  const int segl = min(s0 + col, kSegsPerGemmBlock - 1);  // clamped, tail reads zeros
  const float*  bptr = &sdy[segl * kSegLen + khalf];
  const float2* aptr = &scoA2[(khalf >> 1) * 16 + col];

  v8f acc = {};
#pragma unroll 5
  for (int k0 = 0; k0 < kSegLen; k0 += 4) {
    // A 16x4 fp32: lane row = lane&15; v0 -> K=k0+khalf, v1 -> K=k0+khalf+1
    const float2 av = aptr[(k0 >> 1) * 16];   // one ds_load_b64, imm offset per iter
    v2f aa; aa.x = av.x; aa.y = av.y;
    // B 4x16 fp32: column = lane&15, K rows striped across lane halves
    const float2 bv = *(const float2*)(bptr + k0);
    v2f bb; bb.x = bv.x; bb.y = bv.y;
    acc = __builtin_amdgcn_wmma_f32_16x16x4_f32(
        /*neg_a=*/false, aa, /*neg_b=*/false, bb,
        /*c_mod=*/(short)0, acc, /*reuse_a=*/false, /*reuse_b=*/false);
  }
  // D layout: lanes 0..15 -> N=lane, VGPR r -> M=r. Rows 0,1 are R.
  if (lane < 16 && (s0 + lane) < nsegHere) {
    R[segBase + s0 + lane] = make_float2(acc[0], acc[1]);
  }
}

// ---------------------------------------------------------------------------
// Phase 2: hierarchical boundary scan. 40 threads Horner-accumulate their
// block's 125 segment responses with Mseg = (TL upper-left); thread 0 then
// scans the 40 block transitions with TB, emitting block start states.
// ---------------------------------------------------------------------------
__global__ void k_scan(const Params* __restrict__ P,
                       const float2* __restrict__ R,
                       const float* __restrict__ f0in,
                       double* __restrict__ zb) {
  __shared__ double RB[kNBlocks][2];
  const int tid = threadIdx.x;
  const double m00 = P->TL[0], m01 = P->TL[1], m10 = P->TL[4], m11 = P->TL[5];
  if (tid < kNBlocks) {
    double v0 = 0.0, v1 = 0.0;
    const float2* Rb = R + tid * kSegsPerBlock;
    for (int j = 0; j < kSegsPerBlock; ++j) {
      const double r0 = (double)Rb[j].x, r1 = (double)Rb[j].y;
      const double n0 = m00 * v0 + m01 * v1 + r0;
      const double n1 = m10 * v0 + m11 * v1 + r1;
      v0 = n0; v1 = n1;
    }
    RB[tid][0] = v0; RB[tid][1] = v1;
  }
  __syncthreads();
  if (tid == 0) {
    double TB[16];
    for (int i = 0; i < 16; ++i) TB[i] = P->TB[i];
    double z[4] = {0.0, 0.0, (double)f0in[0], (double)f0in[1]};
    for (int b = 0; b < kNBlocks; ++b) {
      zb[b*4+0] = z[0]; zb[b*4+1] = z[1]; zb[b*4+2] = z[2]; zb[b*4+3] = z[3];
      double nz[4];
      for (int r = 0; r < 4; ++r)
        nz[r] = TB[r*4+0]*z[0] + TB[r*4+1]*z[1] + TB[r*4+2]*z[2] + TB[r*4+3]*z[3];
      nz[0] += RB[b][0]; nz[1] += RB[b][1];
      z[0] = nz[0]; z[1] = nz[1]; z[2] = nz[2]; z[3] = nz[3];
    }
  }
}

// ---------------------------------------------------------------------------
// Phase 3: replay. Thread 0 of each block expands its 125 segment-start
// states (double, via TL + R); then 125 threads each replay 100 f32 steps,
// streaming out xs_hat / dys_hat / fs_hat rows.
// ---------------------------------------------------------------------------
__global__ void __launch_bounds__(128)
k_replay(const float* __restrict__ dys,
         const float* __restrict__ f0in,
         const Params* __restrict__ P,
         const float2* __restrict__ R,
         const double* __restrict__ zb,
         float* __restrict__ xs,
         float* __restrict__ dyh,
         float* __restrict__ fs) {
  __shared__ double zs[kSegsPerBlock][4];
  const int tid = threadIdx.x, b = blockIdx.x;
  if (tid == 0) {
    double TL[16];
    for (int i = 0; i < 16; ++i) TL[i] = P->TL[i];
    double z[4] = {zb[b*4], zb[b*4+1], zb[b*4+2], zb[b*4+3]};
    const float2* Rb = R + b * kSegsPerBlock;
    for (int i = 0; i < kSegsPerBlock; ++i) {
      zs[i][0] = z[0]; zs[i][1] = z[1]; zs[i][2] = z[2]; zs[i][3] = z[3];
      double nz[4];
      for (int r = 0; r < 4; ++r)
        nz[r] = TL[r*4+0]*z[0] + TL[r*4+1]*z[1] + TL[r*4+2]*z[2] + TL[r*4+3]*z[3];
      nz[0] += (double)Rb[i].x; nz[1] += (double)Rb[i].y;
      z[0] = nz[0]; z[1] = nz[1]; z[2] = nz[2]; z[3] = nz[3];
    }
  }
  __syncthreads();
  if (tid >= kSegsPerBlock) return;

  const int seg = b * kSegsPerBlock + tid;
  const long u0 = (long)seg * kSegLen;
  __builtin_prefetch(&dys[2 * u0], 0, 1);   // global_prefetch_b8
  float x0 = (float)zs[tid][0], x1 = (float)zs[tid][1];
  float fa = (float)zs[tid][2], fb = (float)zs[tid][3];
  const float M00 = P->M[0], M01 = P->M[1], M10 = P->M[2], M11 = P->M[3];
  const float F00 = P->F[0], F01 = P->F[1], F10 = P->F[2], F11 = P->F[3];
  const float g0 = P->g[0], g1 = P->g[1];
  const float vx = P->cov[0], vp = P->cov[1], cxp = P->cov[2];
  const float sdt = P->sdt;

  for (int j = 0; j < kSegLen; ++j) {
    const long u = u0 + j;
    const float dy0 = dys[2 * u];
    dyh[2*u]   = sdt * x0;   // C x dt, second component exactly zero
    dyh[2*u+1] = 0.0f;
    const float nf0 = F00 * fa + F01 * fb;
    const float nf1 = F10 * fa + F11 * fb;
    const float nx0 = M00 * x0 + M01 * x1 + g0 * dy0;
    const float nx1 = M10 * x0 + M11 * x1 + g1 * dy0 + kDt * nf0;
    const long r = u + 1;
    float* xr = xs + r * 6;
    xr[0] = nx0; xr[1] = nx1; xr[2] = vx; xr[3] = vp; xr[4] = cxp;
    xr[5] = (float)((double)r * 0.001);
    fs[2*r] = nf0; fs[2*r+1] = nf1;
    x0 = nx0; x1 = nx1; fa = nf0; fb = nf1;
  }
  if (b == 0 && tid == 0) {
    xs[0] = 0.0f; xs[1] = 0.0f; xs[2] = vx; xs[3] = vp; xs[4] = cxp; xs[5] = 0.0f;
    fs[0] = f0in[0]; fs[1] = f0in[1];
  }
}

extern "C" void kernel_launch(void* const* d_in, const int* in_sizes, int n_in,
                              void* d_out, int out_size, void* d_ws, size_t ws_size,
                              hipStream_t stream) {
  (void)in_sizes; (void)n_in; (void)out_size; (void)ws_size;
  const float* dys = (const float*)d_in[0];
  const float* f0  = (const float*)d_in[1];
  const float* Kin = (const float*)d_in[2];

  char* ws = (char*)d_ws;
  Params* P     = (Params*)(ws + 0);
  float2* coefs = (float2*)(ws + 512);
  float2* R     = (float2*)(ws + 2048);                  // 5000 * 8B
  double* zb    = (double*)(ws + 43008);                 // 40 * 4 doubles

  float* xs  = (float*)d_out;                            // (T+1) x 6
  float* dyh = xs  + (size_t)(kTSteps + 1) * 6;          // T x 2
  float* fs  = dyh + (size_t)kTSteps * 2;                // (T+1) x 2

  hipLaunchKernelGGL(k_setup,  dim3(1),           dim3(32),  0, stream, Kin, P, coefs);
  hipLaunchKernelGGL(k_resp,   dim3(kGemmBlocks), dim3(128), 0, stream, dys, coefs, R);
  hipLaunchKernelGGL(k_scan,   dim3(1),           dim3(64),  0, stream, P, R, f0, zb);
  hipLaunchKernelGGL(k_replay, dim3(kNBlocks),    dim3(128), 0, stream, dys, f0, P, R, zb,
                     xs, dyh, fs);
}